// GCNConvolutionGNN_1357209666176
// MI455X (gfx1250) — compile-verified
//
#include <hip/hip_runtime.h>

typedef __attribute__((ext_vector_type(2))) float v2f;
typedef __attribute__((ext_vector_type(4))) float f4;
typedef __attribute__((ext_vector_type(8))) float v8f;

#define HIDDEN  128
#define N_NODES 50000
#define N_EDGES 800000

// ---------------------------------------------------------------------------
// out[16-row stripe] = relu(A @ W + bias) (+ resid) using V_WMMA_F32_16X16X4_F32.
// Block = 256 threads = 8 wave32s; each wave computes one 16x16 tile of the
// 16x128 output stripe. K=128 -> 32 WMMA steps of K=4.
// A-stripe (16x128 = 8KB) is cooperatively staged in LDS; W read via L1/L2.
// ---------------------------------------------------------------------------
template <bool ADD_RESID>
__global__ __launch_bounds__(256)
void gemm_relu_kernel(const float* __restrict__ A,
                      const float* __restrict__ W,
                      const float* __restrict__ bias,
                      const float* __restrict__ resid,
                      float* __restrict__ out) {
    __shared__ float Ald[16 * HIDDEN];

    const int rowBase = blockIdx.x * 16;
    const int tid     = threadIdx.x;

    // Cooperative load of the 16x128 A stripe: 2048 floats = 512 float4s.
    const f4* Ag  = (const f4*)(A + (size_t)rowBase * HIDDEN);
    f4*       As4 = (f4*)Ald;
    As4[tid]       = Ag[tid];
    As4[tid + 256] = Ag[tid + 256];
    __syncthreads();

    const int wave = tid >> 5;          // 0..7 -> 16-column block of output
    const int lane = tid & 31;
    const int m    = lane & 15;         // row within tile (A) / col within tile (B,C,D)
    const int kh   = lane >> 4;         // K-half select per ISA A/B layout
    const int ncol = wave * 16 + m;     // output column 0..127

    // C init: every C/D VGPR r holds column ncol for this lane -> bias broadcast.
    const float bv = bias[ncol];
    v8f c;
#pragma unroll
    for (int r = 0; r < 8; ++r) c[r] = bv;

#pragma unroll 4
    for (int ks = 0; ks < 32; ++ks) {
        const int kk = ks * 4 + kh * 2;           // 8B-aligned (kk even)
        // A frag: lanes 0-15 hold K={kk,kk+1} of row m; lanes 16-31 K={kk,kk+1}(+2 via kh)
        v2f a = *(const v2f*)(&Ald[m * HIDDEN + kk]);
        // B frag: VGPR0 = row K=kk, VGPR1 = row K=kk+1, col ncol
        v2f b;
        b.x = W[(size_t)(kk + 0) * HIDDEN + ncol];
        b.y = W[(size_t)(kk + 1) * HIDDEN + ncol];
        c = __builtin_amdgcn_wmma_f32_16x16x4_f32(
                /*neg_a=*/false, a, /*neg_b=*/false, b,
                /*c_mod=*/(short)0, c, /*reuse_a=*/false, /*reuse_b=*/false);
    }

    // D layout: VGPR r -> (lanes 0-15: M=r, lanes 16-31: M=r+8), N = lane&15.
#pragma unroll
    for (int r = 0; r < 8; ++r) {
        const int    row = rowBase + r + kh * 8;
        const size_t idx = (size_t)row * HIDDEN + ncol;
        float v = c[r];
        v = v > 0.0f ? v : 0.0f;
        if (ADD_RESID) v += resid[idx];
        out[idx] = v;
    }
}

// ---------------------------------------------------------------------------
// pooled[tgt[e]] += norm[e] * H1[src[e]] : one wave32 per edge, float4/lane.
// H1 (25.6MB) and pooled (25.6MB) both live in the 192MB L2 -> atomics in L2.
// ---------------------------------------------------------------------------
__global__ __launch_bounds__(256)
void scatter_kernel(const float* __restrict__ H1,
                    const int* __restrict__ src,
                    const int* __restrict__ tgt,
                    const float* __restrict__ norm,
                    float* __restrict__ pooled) {
    const int gid  = blockIdx.x * blockDim.x + threadIdx.x;
    const int e    = gid >> 5;
    const int lane = gid & 31;
    if (e >= N_EDGES) return;

    const int   s = src[e];
    const int   t = tgt[e];
    const float w = norm[e];

    const f4 v = *(const f4*)(H1 + (size_t)s * HIDDEN + lane * 4);
    float* dst = pooled + (size_t)t * HIDDEN + lane * 4;
    atomicAdd(dst + 0, w * v.x);
    atomicAdd(dst + 1, w * v.y);
    atomicAdd(dst + 2, w * v.z);
    atomicAdd(dst + 3, w * v.w);
}

__global__ __launch_bounds__(256)
void zero_kernel(float* __restrict__ p, int n4) {
    const int i = blockIdx.x * blockDim.x + threadIdx.x;
    if (i < n4) ((f4*)p)[i] = f4{0.0f, 0.0f, 0.0f, 0.0f};
}

// ---------------------------------------------------------------------------
extern "C" void kernel_launch(void* const* d_in, const int* in_sizes, int n_in,
                              void* d_out, int out_size, void* d_ws, size_t ws_size,
                              hipStream_t stream) {
    const float* X    = (const float*)d_in[0];   // node_features [N,H]
    const int*   src  = (const int*)  d_in[1];   // [E]
    const int*   tgt  = (const int*)  d_in[2];   // [E]
    const float* norm = (const float*)d_in[3];   // [E]
    const float* W1   = (const float*)d_in[4];   // [H,H]
    const float* b1   = (const float*)d_in[5];   // [H]
    const float* W2   = (const float*)d_in[6];   // [H,H]
    const float* b2   = (const float*)d_in[7];   // [H]

    float* out    = (float*)d_out;               // [N,H]
    float* pooled = (float*)d_ws;                // [N,H] scratch (25.6 MB)
    float* H1     = out;                         // stage relu(X@W1+b1) in d_out

    // 1) per-node hidden: H1 = relu(X @ W1 + b1)   (16x FLOP cut vs per-edge)
    gemm_relu_kernel<false><<<N_NODES / 16, 256, 0, stream>>>(X, W1, b1, nullptr, H1);

    // 2) pooled = 0
    zero_kernel<<<(N_NODES * HIDDEN / 4 + 255) / 256, 256, 0, stream>>>(
        pooled, N_NODES * HIDDEN / 4);

    // 3) pooled[tgt] += norm * H1[src]   (one wave per edge; E/8 blocks exactly)
    scatter_kernel<<<N_EDGES / 8, 256, 0, stream>>>(H1, src, tgt, norm, pooled);

    // 4) out = relu(pooled @ W2 + b2) + X
    gemm_relu_kernel<true><<<N_NODES / 16, 256, 0, stream>>>(pooled, W2, b2, X, out);
}